// Count_41506563948881
// MI455X (gfx1250) — compile-verified
//
#include <hip/hip_runtime.h>
#include <hip/hip_bf16.h>
#include <stdint.h>

// Bilinear splat ("count") through a displacement field, wrap boundary.
// phi: (B, 2, H, W) float32 -> out: (B, 1, H, W) float32
//
// MI455X strategy:
//  - 256-thread block owns a 64x64 output tile; accumulates all 4 bilinear
//    taps per pixel into an 80x80 LDS tile (64 + 2*8 halo) via ds_add_f32.
//  - float4 (global_load_b128) reads of both displacement planes.
//  - Flush + rare Gaussian-tail fallback use explicit inline-asm
//    global_atomic_add_f32 at device scope (neighbor tiles' halos overlap,
//    so WGP scope would be incorrect). Output (67MB) is L2-resident (192MB).
//  - Output zero-initialized first by a float4-store kernel.

#define TH    64
#define TW    64
#define HALO  8
#define LH    (TH + 2 * HALO)   // 80
#define LW    (TW + 2 * HALO)   // 80
#define NTHR  256

__device__ __forceinline__ void gfadd(float* p, float v) {
    // Non-returning FP32 atomic add, device scope (cross-workgroup visible).
    asm volatile("global_atomic_add_f32 %0, %1, off scope:SCOPE_DEV"
                 : : "v"(p), "v"(v) : "memory");
}

__device__ __forceinline__ int wrap_full(int i, int n) {
    int m = i % n;
    return (m < 0) ? m + n : m;
}

// halo coords lie in [-HALO, n+HALO): one conditional fix-up suffices
__device__ __forceinline__ int wrap_near(int i, int n) {
    if (i < 0)  return i + n;
    if (i >= n) return i - n;
    return i;
}

__global__ __launch_bounds__(NTHR) void zero_f4_kernel(float4* __restrict__ out,
                                                       long long n4) {
    long long i = (long long)blockIdx.x * NTHR + threadIdx.x;
    if (i < n4) out[i] = make_float4(0.f, 0.f, 0.f, 0.f);
}

__device__ __forceinline__ void splat_pixel(float* __restrict__ tile,
                                            float* __restrict__ outb,
                                            int gi, int gj, float dx, float dy,
                                            int lrow0, int lcol0, int H, int W) {
    const float x   = (float)gi + dx;
    const float y   = (float)gj + dy;
    const float x0f = floorf(x);
    const float y0f = floorf(y);
    const float wx1 = x - x0f;
    const float wy1 = y - y0f;
    const float wx0 = 1.0f - wx1;
    const float wy0 = 1.0f - wy1;

    const int ix0 = (int)x0f;            // unwrapped target row
    const int iy0 = (int)y0f;            // unwrapped target col
    const int lx  = ix0 - lrow0;         // LDS-local coords
    const int ly  = iy0 - lcol0;

    if ((unsigned)lx < (unsigned)(LH - 1) && (unsigned)ly < (unsigned)(LW - 1)) {
        // hot path: all 4 taps inside LDS tile -> ds_add_f32 (fire & forget)
        float* p = &tile[lx * LW + ly];
        atomicAdd(p,          wx0 * wy0);
        atomicAdd(p + 1,      wx0 * wy1);
        atomicAdd(p + LW,     wx1 * wy0);
        atomicAdd(p + LW + 1, wx1 * wy1);
    } else {
        // Gaussian-tail fallback (~1e-9 / component): direct global atomics
        const int a0 = wrap_full(ix0, H);
        const int a1 = (a0 + 1 == H) ? 0 : a0 + 1;
        const int b0 = wrap_full(iy0, W);
        const int b1 = (b0 + 1 == W) ? 0 : b0 + 1;
        gfadd(&outb[(size_t)a0 * W + b0], wx0 * wy0);
        gfadd(&outb[(size_t)a0 * W + b1], wx0 * wy1);
        gfadd(&outb[(size_t)a1 * W + b0], wx1 * wy0);
        gfadd(&outb[(size_t)a1 * W + b1], wx1 * wy1);
    }
}

__global__ __launch_bounds__(NTHR) void splat_kernel(const float* __restrict__ phi,
                                                     float* __restrict__ out,
                                                     int B, int H, int W) {
    __shared__ float tile[LH * LW];

    // zero the LDS accumulator tile
    for (int i = threadIdx.x; i < LH * LW; i += NTHR) tile[i] = 0.0f;
    __syncthreads();

    const int tilesW = W / TW;
    const int tilesH = H / TH;
    const int tilesPerImg = tilesW * tilesH;

    int bid = blockIdx.x;
    int b   = bid / tilesPerImg;
    int t   = bid - b * tilesPerImg;
    int ty  = t / tilesW;
    int tx  = t - ty * tilesW;

    const int row0  = ty * TH;          // tile origin in the image
    const int col0  = tx * TW;
    const int lrow0 = row0 - HALO;      // LDS origin in unwrapped coords
    const int lcol0 = col0 - HALO;

    const size_t plane = (size_t)H * W;
    const float* __restrict__ dxp = phi + (size_t)b * 2 * plane;  // phi[:,0]
    const float* __restrict__ dyp = dxp + plane;                  // phi[:,1]
    float* __restrict__ outb = out + (size_t)b * plane;

    // 256 threads: 16 rows x (16 threads * 4 cols) per step; 4 steps = 64 rows.
    // Each thread does one float4 load per plane per step (global_load_b128).
    const int cg = (threadIdx.x & 15) << 2;   // col offset in tile: 0,4,...,60
    const int r0 = threadIdx.x >> 4;          // 0..15

    for (int rr = r0; rr < TH; rr += NTHR / 16) {
        const int gi = row0 + rr;
        const size_t off = (size_t)gi * W + (col0 + cg);

        const float4 dx4 = *(const float4*)(dxp + off);
        const float4 dy4 = *(const float4*)(dyp + off);

        const int gj = col0 + cg;
        splat_pixel(tile, outb, gi, gj + 0, dx4.x, dy4.x, lrow0, lcol0, H, W);
        splat_pixel(tile, outb, gi, gj + 1, dx4.y, dy4.y, lrow0, lcol0, H, W);
        splat_pixel(tile, outb, gi, gj + 2, dx4.z, dy4.z, lrow0, lcol0, H, W);
        splat_pixel(tile, outb, gi, gj + 3, dx4.w, dy4.w, lrow0, lcol0, H, W);
    }
    __syncthreads();

    // flush LDS tile to global (atomic: neighbor tiles' halos overlap us)
    for (int i = threadIdx.x; i < LH * LW; i += NTHR) {
        const float v = tile[i];
        if (v != 0.0f) {
            const int lr = i / LW;
            const int lc = i - lr * LW;
            const int gr = wrap_near(lr + lrow0, H);
            const int gc = wrap_near(lc + lcol0, W);
            gfadd(&outb[(size_t)gr * W + gc], v);
        }
    }
}

extern "C" void kernel_launch(void* const* d_in, const int* in_sizes, int n_in,
                              void* d_out, int out_size, void* d_ws, size_t ws_size,
                              hipStream_t stream) {
    const float* phi = (const float*)d_in[0];
    float* out = (float*)d_out;

    const int H = 2048, W = 2048;
    const int B = out_size / (H * W);            // out_size = B*H*W

    // 1) zero-init the output (harness poisons it)
    long long n4 = (long long)out_size / 4;
    long long zgrid = (n4 + NTHR - 1) / NTHR;
    zero_f4_kernel<<<(unsigned)zgrid, NTHR, 0, stream>>>((float4*)out, n4);

    // 2) tiled splat with LDS accumulation
    int nblocks = B * (H / TH) * (W / TW);       // 4 * 32 * 32 = 4096
    splat_kernel<<<nblocks, NTHR, 0, stream>>>(phi, out, B, H, W);
}